// DigitCaps_60876866453877
// MI455X (gfx1250) — compile-verified
//
#include <hip/hip_runtime.h>
#include <hip/hip_bf16.h>

// Problem constants (from reference setup_inputs)
#define BATCH   512
#define NPRIM   4608   // N primary capsules
#define DIN     16
#define DOUT    32
#define CAPS    32
#define ITERS   3

// Tiling constants
#define NC      128    // n's per S-kernel block
#define NGROUP  16     // n's per LDS staging group
#define KSTRIDE 20     // padded floats per k-row in u-staging (16 b + 4 pad)
#define UROW    644    // padded floats per n in u-staging (32*20 + 4 pad)
#define XB      20     // padded floats per b-row in x-staging (16 i + 4 pad)
#define XROW    320    // floats per n in x-staging (16*20)
#define CROW    33     // padded floats per n-row of routing coefficients

typedef float v2f __attribute__((ext_vector_type(2)));
typedef float v8f __attribute__((ext_vector_type(8)));
typedef int   v4i __attribute__((ext_vector_type(4)));

// CDNA5 async global->LDS path (ASYNCcnt-tracked), if this toolchain has it.
#if __has_builtin(__builtin_amdgcn_global_load_async_to_lds_b128) && \
    __has_builtin(__builtin_amdgcn_s_wait_asynccnt)
#define USE_ASYNC_LDS 1
#else
#define USE_ASYNC_LDS 0
#endif

__device__ __forceinline__ void async_stage_wait() {
#if USE_ASYNC_LDS
  __builtin_amdgcn_s_wait_asynccnt(0);
#endif
}

__device__ __forceinline__ v8f v8f_zero() {
  v8f z;
#pragma unroll
  for (int i = 0; i < 8; ++i) z[i] = 0.0f;
  return z;
}

// D(16x16,f32) = A(16x4,f32) * B(4x16,f32) + C   -> v_wmma_f32_16x16x4_f32
__device__ __forceinline__ v8f wmma_f32(v2f a, v2f b, v8f c) {
  return __builtin_amdgcn_wmma_f32_16x16x4_f32(false, a, false, b, (short)0, c,
                                               false, false);
}

// ---------------------------------------------------------------------------
// Cooperative x staging: 16 b-rows x 16 n's x 16 inputs -> padded LDS tile.
// Thread t covers 64B contiguous. Async variant DMAs straight into LDS
// (global_load_async_to_lds_b128), retired by s_wait_asynccnt at the loop top.
// ---------------------------------------------------------------------------
__device__ __forceinline__ void stage_x(const float* __restrict__ x, float* xs,
                                        int b0, int nbase, int tid) {
  const int b  = tid >> 4;    // 0..15
  const int nl = tid & 15;    // 0..15
  const float* src = x + (size_t)(b0 + b) * (NPRIM * DIN)
                       + (size_t)(nbase + nl) * DIN;
  float* dst = &xs[nl * XROW + b * XB];
#if USE_ASYNC_LDS
#pragma unroll
  for (int q = 0; q < 4; ++q)
    __builtin_amdgcn_global_load_async_to_lds_b128(
        (__attribute__((address_space(1))) v4i*)(src + 4 * q),
        (__attribute__((address_space(3))) v4i*)(dst + 4 * q), 0, 0);
#else
#pragma unroll
  for (int q = 0; q < 4; ++q) ((float4*)dst)[q] = ((const float4*)src)[q];
#endif
}

// ---------------------------------------------------------------------------
// Phase 1 (shared by S and B kernels): each wave computes u_hat for two n's of
// the current 16-n group via 8 WMMAs per n (A = x[16b x 16], B = rw[n] 16x32),
// then stores the C-layout result into LDS as u[nl][k*KSTRIDE + b] with
// aligned ds_store_b128 (consecutive VGPRs of a D tile hold consecutive b).
// ---------------------------------------------------------------------------
__device__ __forceinline__ void uhat_phase(const float* __restrict__ rw,
                                           const float* xs, float* ustg,
                                           int nbase, int wave, int l15, int hi) {
#pragma unroll
  for (int rep = 0; rep < 2; ++rep) {
    const int nl = wave * 2 + rep;
    const float* rwn = rw + (size_t)(nbase + nl) * (DIN * DOUT);
    v8f d0 = v8f_zero(), d1 = v8f_zero();
#pragma unroll
    for (int t = 0; t < 4; ++t) {
      // A chunk 16x4: lanes0-15 K=4t+{0,1}, lanes16-31 K=4t+{2,3}
      v2f a = *(const v2f*)&xs[nl * XROW + l15 * XB + 4 * t + 2 * hi];
      const int r0 = 4 * t + 2 * hi;
      v2f b0, b1;
      b0[0] = rwn[(r0    ) * DOUT + l15];        // cols 0..15
      b0[1] = rwn[(r0 + 1) * DOUT + l15];
      b1[0] = rwn[(r0    ) * DOUT + 16 + l15];   // cols 16..31
      b1[1] = rwn[(r0 + 1) * DOUT + 16 + l15];
      d0 = wmma_f32(a, b0, d0);
      d1 = wmma_f32(a, b1, d1);
    }
    // store D tiles: lane has col k = l15(+16), VGPR j has row b = j + 8*hi
    float4* p0 = (float4*)&ustg[nl * UROW + l15 * KSTRIDE + 8 * hi];
    p0[0] = make_float4(d0[0], d0[1], d0[2], d0[3]);
    p0[1] = make_float4(d0[4], d0[5], d0[6], d0[7]);
    float4* p1 = (float4*)&ustg[nl * UROW + (16 + l15) * KSTRIDE + 8 * hi];
    p1[0] = make_float4(d1[0], d1[1], d1[2], d1[3]);
    p1[1] = make_float4(d1[4], d1[5], d1[6], d1[7]);
  }
}

// ---------------------------------------------------------------------------
// Softmax over the 32 routing logits per primary capsule (wave per row).
// ---------------------------------------------------------------------------
__global__ __launch_bounds__(256) void caps_softmax_kernel(
    const float* __restrict__ blog, float* __restrict__ cmat) {
  const int row  = blockIdx.x * 8 + (threadIdx.x >> 5);
  const int lane = threadIdx.x & 31;
  float v = blog[(size_t)row * CAPS + lane];
  float m = v;
#pragma unroll
  for (int off = 16; off > 0; off >>= 1) m = fmaxf(m, __shfl_xor(m, off, 32));
  float e = __expf(v - m);
  float sum = e;
#pragma unroll
  for (int off = 16; off > 0; off >>= 1) sum += __shfl_xor(sum, off, 32);
  cmat[(size_t)row * CAPS + lane] = e / sum;
}

// ---------------------------------------------------------------------------
// S kernel: fused u_hat (WMMA) + s[b,cap,k] += sum_n u_hat[b,n,k]*c[n,cap].
// The s-contraction is a GEMM per k-slice: rows=b, K=n, cols=cap.
// Wave w keeps accumulators for k in {4w..4w+3} x 2 cap-halves in registers
// across the whole 128-n chunk, then atomically adds to global s.
// x tiles are double-buffered: the next group's async copy overlaps compute.
// ---------------------------------------------------------------------------
__global__ __launch_bounds__(256) void caps_s_kernel(
    const float* __restrict__ x, const float* __restrict__ rw,
    const float* __restrict__ cmat, float* __restrict__ s) {
  __shared__ float ustg[NGROUP * UROW];      // u_hat staging [nl][k*20 + b]
  __shared__ float xs[2][NGROUP * XROW];     // x staging (double-buffered)
  __shared__ float cstg[NC * CROW];          // c staging [nl][cap]

  const int tid  = threadIdx.x;
  const int lane = tid & 31;
  const int wave = tid >> 5;              // 0..7
  const int hi   = (lane >> 4) & 1;
  const int l15  = lane & 15;

  const int b0    = blockIdx.y * 16;
  const int nbase = blockIdx.x * NC;

  for (int idx = tid; idx < NC * CAPS; idx += 256) {
    const int nl = idx >> 5, cap = idx & 31;
    cstg[nl * CROW + cap] = cmat[(size_t)(nbase + nl) * CAPS + cap];
  }

  v8f acc[4][2];
#pragma unroll
  for (int i = 0; i < 4; ++i) {
    acc[i][0] = v8f_zero();
    acc[i][1] = v8f_zero();
  }

  stage_x(x, xs[0], b0, nbase, tid);         // prologue: group 0

  const int G = NC / NGROUP;
  for (int g = 0; g < G; ++g) {
    const int ng = nbase + g * NGROUP;
    async_stage_wait();                      // xs[g&1] landed (async path)
    __syncthreads();                         // publish xs; ustg free again
    if (g + 1 < G) {
      stage_x(x, xs[(g + 1) & 1], b0, ng + NGROUP, tid);   // overlap w/ compute
      // prefetch next group's route-weight slice (32KB = 256 x 128B lines)
      __builtin_prefetch(rw + (size_t)(ng + NGROUP) * (DIN * DOUT) + tid * 32,
                         0, 1);
    }
    uhat_phase(rw, xs[g & 1], ustg, ng, wave, l15, hi);
    __syncthreads();

    // s-GEMM over this group's 16 n's (4 K-steps of 4 n's)
#pragma unroll
    for (int step = 0; step < 4; ++step) {
      const int nlc = g * NGROUP + step * 4;  // chunk-local base n
      const int nlg = step * 4;               // group-local base n
      v2f bc0, bc1;  // B = c[n,cap], shared across k-slices
      bc0[0] = cstg[(nlc + 2 * hi    ) * CROW + l15];
      bc0[1] = cstg[(nlc + 2 * hi + 1) * CROW + l15];
      bc1[0] = cstg[(nlc + 2 * hi    ) * CROW + 16 + l15];
      bc1[1] = cstg[(nlc + 2 * hi + 1) * CROW + 16 + l15];
#pragma unroll
      for (int kl = 0; kl < 4; ++kl) {
        const int k = wave * 4 + kl;
        v2f a;  // A[row=b, K=n] = u_hat[b, n, k]
        a[0] = ustg[(nlg + 2 * hi    ) * UROW + k * KSTRIDE + l15];
        a[1] = ustg[(nlg + 2 * hi + 1) * UROW + k * KSTRIDE + l15];
        acc[kl][0] = wmma_f32(a, bc0, acc[kl][0]);
        acc[kl][1] = wmma_f32(a, bc1, acc[kl][1]);
      }
    }
  }

  // Flush partial s tiles (one of N/NC partials per location)
#pragma unroll
  for (int kl = 0; kl < 4; ++kl) {
    const int k = wave * 4 + kl;
#pragma unroll
    for (int ch = 0; ch < 2; ++ch) {
      const int cap = l15 + 16 * ch;
#pragma unroll
      for (int j = 0; j < 8; ++j) {
        const int b = b0 + j + 8 * hi;
        unsafeAtomicAdd(&s[((size_t)b * CAPS + cap) * DOUT + k], acc[kl][ch][j]);
      }
    }
  }
}

// ---------------------------------------------------------------------------
// Squash: v = (|s|^2 / (1+|s|^2)) * s / |s|  over the 32-wide last axis.
// ---------------------------------------------------------------------------
__global__ __launch_bounds__(256) void caps_squash_kernel(
    const float* __restrict__ s, float* __restrict__ v) {
  const int row  = blockIdx.x * 8 + (threadIdx.x >> 5);   // b*CAPS + cap
  const int lane = threadIdx.x & 31;
  const float val = s[(size_t)row * DOUT + lane];
  float sq = val * val;
#pragma unroll
  for (int off = 16; off > 0; off >>= 1) sq += __shfl_xor(sq, off, 32);
  const float scale = sq / ((1.0f + sq) * __fsqrt_rn(sq));
  v[(size_t)row * DOUT + lane] = val * scale;
}

// ---------------------------------------------------------------------------
// Agreement kernel: blog[n,cap] += sum_{b,k} u_hat[b,n,k] * v[b,cap,k].
// GEMM rows = n (16 per block), K = (k,b) (512 per b-tile), cols = cap.
// u_hat is recomputed per b-tile with WMMA and staged through LDS; each wave
// owns 4 of the 32 k (an eighth of K); partials reduced via LDS float atomics.
// x tiles double-buffered across the b-tile loop like the S kernel.
// ---------------------------------------------------------------------------
__global__ __launch_bounds__(256) void caps_b_kernel(
    const float* __restrict__ x, const float* __restrict__ rw,
    const float* __restrict__ v, float* __restrict__ blog) {
  __shared__ float ustg[NGROUP * UROW];
  __shared__ float xs[2][NGROUP * XROW];
  __shared__ float dblds[16 * CAPS];

  const int tid  = threadIdx.x;
  const int lane = tid & 31;
  const int wave = tid >> 5;
  const int hi   = (lane >> 4) & 1;
  const int l15  = lane & 15;

  const int n0 = blockIdx.x * 16;

  for (int i = tid; i < 16 * CAPS; i += 256) dblds[i] = 0.0f;

  v8f acc0 = v8f_zero(), acc1 = v8f_zero();

  stage_x(x, xs[0], 0, n0, tid);             // prologue: b-tile 0

  for (int bt = 0; bt < BATCH / 16; ++bt) {
    const int b0 = bt * 16;
    async_stage_wait();
    __syncthreads();
    if (bt + 1 < BATCH / 16)
      stage_x(x, xs[(bt + 1) & 1], b0 + 16, n0, tid);      // overlap w/ compute
    uhat_phase(rw, xs[bt & 1], ustg, n0, wave, l15, hi);
    __syncthreads();

#pragma unroll
    for (int kl = 0; kl < 4; ++kl) {
      const int k = wave * 4 + kl;
#pragma unroll
      for (int bb = 0; bb < 16; bb += 4) {
        // A[row=n, K=k*16+b] = u_hat[b, n0+row, k]  (contiguous b pair -> b64)
        v2f a = *(const v2f*)&ustg[l15 * UROW + k * KSTRIDE + bb + 2 * hi];
        const int brow = b0 + bb + 2 * hi;
        v2f bv0, bv1;  // B[K, cap] = v[b, cap, k]
        bv0[0] = v[((size_t)(brow    ) * CAPS + l15) * DOUT + k];
        bv0[1] = v[((size_t)(brow + 1) * CAPS + l15) * DOUT + k];
        bv1[0] = v[((size_t)(brow    ) * CAPS + 16 + l15) * DOUT + k];
        bv1[1] = v[((size_t)(brow + 1) * CAPS + 16 + l15) * DOUT + k];
        acc0 = wmma_f32(a, bv0, acc0);
        acc1 = wmma_f32(a, bv1, acc1);
      }
    }
  }

  __syncthreads();
  // Cross-wave reduction of k-partials: D rows = n_local = j + 8*hi
#pragma unroll
  for (int j = 0; j < 8; ++j) {
    const int nl = j + 8 * hi;
    atomicAdd(&dblds[nl * CAPS + l15],      acc0[j]);
    atomicAdd(&dblds[nl * CAPS + 16 + l15], acc1[j]);
  }
  __syncthreads();
  for (int i = tid; i < 16 * CAPS; i += 256) {
    const int nl = i >> 5, cap = i & 31;
    blog[(size_t)(n0 + nl) * CAPS + cap] += dblds[i];  // block owns these rows
  }
}

// ---------------------------------------------------------------------------
extern "C" void kernel_launch(void* const* d_in, const int* in_sizes, int n_in,
                              void* d_out, int out_size, void* d_ws,
                              size_t ws_size, hipStream_t stream) {
  (void)in_sizes; (void)n_in; (void)out_size; (void)ws_size;
  const float* x  = (const float*)d_in[0];
  const float* rw = (const float*)d_in[1];

  float* s    = (float*)d_ws;                          // 512*32*32 f32 (2 MB)
  float* blog = s + (size_t)BATCH * CAPS * DOUT;       // 4608*32 f32
  float* cmat = blog + (size_t)NPRIM * CAPS;           // 4608*32 f32
  float* v    = (float*)d_out;                         // 512*32*32 f32

  (void)hipMemsetAsync(blog, 0, sizeof(float) * (size_t)NPRIM * CAPS, stream);

  for (int it = 0; it < ITERS; ++it) {
    caps_softmax_kernel<<<NPRIM / 8, 256, 0, stream>>>(blog, cmat);
    (void)hipMemsetAsync(s, 0, sizeof(float) * (size_t)BATCH * CAPS * DOUT,
                         stream);
    caps_s_kernel<<<dim3(NPRIM / NC, BATCH / 16), 256, 0, stream>>>(x, rw, cmat, s);
    caps_squash_kernel<<<(BATCH * CAPS) / 8, 256, 0, stream>>>(s, v);
    if (it + 1 < ITERS)
      caps_b_kernel<<<NPRIM / 16, 256, 0, stream>>>(x, rw, v, blog);
  }
}